// NonAutoregressiveDecoder_83150566851200
// MI455X (gfx1250) — compile-verified
//
#include <hip/hip_runtime.h>
#include <hip/hip_bf16.h>

// ---------------- problem constants ----------------
constexpr int B_    = 32;
constexpr int L_    = 512;
constexpr int DENC  = 512;
constexpr int S_    = 32;
constexpr int E_    = 128;
constexpr int A_    = 256;
constexpr int H_    = 512;
constexpr int V_    = 32000;
constexpr int M1    = B_ * L_;   // 16384 rows for enc_proj
constexpr int M3    = B_ * S_;   // 1024 rows for logits

typedef float v2f __attribute__((ext_vector_type(2)));
typedef float v8f __attribute__((ext_vector_type(8)));

// Full-precision CDNA5 WMMA: D(16x16,f32) = A(16x4,f32) * B(4x16,f32) + C
__device__ __forceinline__ v8f wmma_f32(v2f a, v2f b, v8f c) {
    return __builtin_amdgcn_wmma_f32_16x16x4_f32(
        /*neg_a=*/false, a, /*neg_b=*/false, b,
        /*c_mod=*/(short)0, c, /*reuse_a=*/false, /*reuse_b=*/false);
}

__device__ __forceinline__ float tanh_fast(float x) {
    // v_exp_f32-based tanh; exact limits at +-inf
    float e = __expf(2.0f * x);
    return 1.0f - 2.0f / (e + 1.0f);
}

// CDNA5 async global->LDS copy (ASYNCcnt-tracked), 16B per lane.
// lds_off = LDS byte offset (low 32 bits of a generic __shared__ pointer).
__device__ __forceinline__ void async_b128(unsigned lds_off, const void* gptr) {
    asm volatile("global_load_async_to_lds_b128 %0, %1, off"
                 :: "v"(lds_off), "v"(gptr) : "memory");
}
__device__ __forceinline__ void wait_async0() {
    asm volatile("s_wait_asynccnt 0" ::: "memory");
}

// ---------------------------------------------------------------
// K0: dec_query[s,a] = pos[s,:]·dec_proj_W[a,:] + db[a]   (32x256)
//     hidden[s,h]    = pos[s,:]·hidden_proj_W[h,:] + hb[h] (32x512)
// ---------------------------------------------------------------
__global__ __launch_bounds__(256) void small_proj_kernel(
    const float* __restrict__ pos, const float* __restrict__ dW,
    const float* __restrict__ db,  const float* __restrict__ hW,
    const float* __restrict__ hb,  float* __restrict__ dq,
    float* __restrict__ hid)
{
    int g = blockIdx.x * 256 + threadIdx.x;
    if (g < S_ * A_) {
        int s = g >> 8, a = g & (A_ - 1);
        const float* p = pos + s * E_;
        const float* w = dW + a * E_;
        float acc = db[a];
#pragma unroll 8
        for (int k = 0; k < E_; ++k) acc = fmaf(p[k], w[k], acc);
        dq[g] = acc;
    } else {
        g -= S_ * A_;
        if (g < S_ * H_) {
            int s = g >> 9, h = g & (H_ - 1);
            const float* p = pos + s * E_;
            const float* w = hW + h * E_;
            float acc = hb[h];
#pragma unroll 8
            for (int k = 0; k < E_; ++k) acc = fmaf(p[k], w[k], acc);
            hid[g] = acc;
        }
    }
}

// ---------------------------------------------------------------
// K1: enc_proj[m,n] = X[m,:]·W[n,:] + b[n]   (16384x256, K=512)
//     one 16x16 tile per wave, f32 WMMA, K step 4
// ---------------------------------------------------------------
__global__ __launch_bounds__(256) void enc_proj_kernel(
    const float* __restrict__ X, const float* __restrict__ W,
    const float* __restrict__ bvec, float* __restrict__ out)
{
    const int wave  = blockIdx.x * 8 + (threadIdx.x >> 5);
    const int nTile = wave & 15;       // 256/16 tiles
    const int mTile = wave >> 4;       // 1024 tiles
    const int t = threadIdx.x & 31;
    const int half = t >> 4, mr = t & 15;
    const float* arow = X + (size_t)(mTile * 16 + mr) * DENC;
    const float* wrow = W + (size_t)(nTile * 16 + mr) * DENC;
    v8f acc = {};
#pragma unroll 4
    for (int k0 = 0; k0 < DENC; k0 += 4) {
        v2f a = *(const v2f*)(arow + k0 + 2 * half);
        v2f b = *(const v2f*)(wrow + k0 + 2 * half);
        acc = wmma_f32(a, b, acc);
    }
    const float bn = bvec[nTile * 16 + mr];
#pragma unroll
    for (int j = 0; j < 8; ++j) {
        int m = mTile * 16 + half * 8 + j;
        out[(size_t)m * A_ + nTile * 16 + mr] = acc[j] + bn;
    }
}

// ---------------------------------------------------------------
// K2a: scores[b,s,l] = sum_a tanh(encp[b,l,a] + dq[s,a]) * w[a] + b0
//      One thread owns one l and accumulates ALL 32 steps in regs,
//      so enc_proj is read exactly once (16 MB total, not 512 MB).
//      dq table (32x256 = 32 KB) + score weights live in LDS.
// ---------------------------------------------------------------
__global__ __launch_bounds__(256) void score_kernel(
    const float* __restrict__ encp, const float* __restrict__ dq,
    const float* __restrict__ w,    const float* __restrict__ b0,
    float* __restrict__ scores)
{
    __shared__ float dqs[S_ * A_];   // 32 KB
    __shared__ float wsh[A_];

    const int b  = blockIdx.x >> 1;          // 32
    const int lc = blockIdx.x & 1;           // L split in 2 chunks of 256
    const int tid = threadIdx.x;
    const int l = lc * 256 + tid;

#pragma unroll
    for (int i = 0; i < S_ * A_ / 256; ++i) dqs[tid + i * 256] = dq[tid + i * 256];
    wsh[tid] = w[tid];
    __syncthreads();

    const float* erow = encp + ((size_t)b * L_ + l) * A_;
    float acc[S_];
#pragma unroll
    for (int s = 0; s < S_; ++s) acc[s] = 0.0f;

    for (int a0 = 0; a0 < A_; a0 += 8) {
        const float4 e0 = *(const float4*)(erow + a0);
        const float4 e1 = *(const float4*)(erow + a0 + 4);
        const float4 w0 = *(const float4*)(wsh + a0);
        const float4 w1 = *(const float4*)(wsh + a0 + 4);
#pragma unroll
        for (int s = 0; s < S_; ++s) {
            const float4 d0 = *(const float4*)(dqs + s * A_ + a0);
            const float4 d1 = *(const float4*)(dqs + s * A_ + a0 + 4);
            float sum = acc[s];
            sum = fmaf(tanh_fast(e0.x + d0.x), w0.x, sum);
            sum = fmaf(tanh_fast(e0.y + d0.y), w0.y, sum);
            sum = fmaf(tanh_fast(e0.z + d0.z), w0.z, sum);
            sum = fmaf(tanh_fast(e0.w + d0.w), w0.w, sum);
            sum = fmaf(tanh_fast(e1.x + d1.x), w1.x, sum);
            sum = fmaf(tanh_fast(e1.y + d1.y), w1.y, sum);
            sum = fmaf(tanh_fast(e1.z + d1.z), w1.z, sum);
            sum = fmaf(tanh_fast(e1.w + d1.w), w1.w, sum);
            acc[s] = sum;
        }
    }
    const float scb = b0[0];
#pragma unroll
    for (int s = 0; s < S_; ++s)
        scores[((size_t)b * S_ + s) * L_ + l] = acc[s] + scb;
}

// ---------------------------------------------------------------
// K2b: in-place softmax over l for each (b,s) row of 512
// ---------------------------------------------------------------
__global__ __launch_bounds__(256) void softmax_kernel(float* __restrict__ attn)
{
    __shared__ float red[256];
    const int tid = threadIdx.x;
    float* row = attn + (size_t)blockIdx.x * L_;

    float v0 = row[tid], v1 = row[tid + 256];
    red[tid] = fmaxf(v0, v1);
    __syncthreads();
    for (int st = 128; st > 0; st >>= 1) {
        if (tid < st) red[tid] = fmaxf(red[tid], red[tid + st]);
        __syncthreads();
    }
    const float mx = red[0];
    __syncthreads();

    float e0 = __expf(v0 - mx);
    float e1 = __expf(v1 - mx);
    red[tid] = e0 + e1;
    __syncthreads();
    for (int st = 128; st > 0; st >>= 1) {
        if (tid < st) red[tid] += red[tid + st];
        __syncthreads();
    }
    const float inv = 1.0f / red[0];
    row[tid]       = e0 * inv;
    row[tid + 256] = e1 * inv;
}

// ---------------------------------------------------------------
// K3: context[b,s,d] = sum_l attn[b,s,l] * enc_out[b,l,d]
//     batched [32x512]·[512x512], one 16x16 tile per wave
// ---------------------------------------------------------------
__global__ __launch_bounds__(256) void context_kernel(
    const float* __restrict__ AW, const float* __restrict__ ENC,
    float* __restrict__ ctx)
{
    const int wave  = blockIdx.x * 8 + (threadIdx.x >> 5);
    const int nTile = wave & 31;        // 512/16
    const int rest  = wave >> 5;
    const int mTile = rest & 1;         // 32 rows -> 2 tiles
    const int b     = rest >> 1;        // 32 batches
    const int t = threadIdx.x & 31;
    const int half = t >> 4, mr = t & 15;
    const int n0 = nTile * 16;

    const float* arow = AW  + ((size_t)b * S_ + mTile * 16 + mr) * L_;
    const float* Bb   = ENC + (size_t)b * L_ * DENC;
    v8f acc = {};
    for (int k0 = 0; k0 < L_; k0 += 4) {
        v2f a = *(const v2f*)(arow + k0 + 2 * half);
        v2f bb;
        bb.x = Bb[(size_t)(k0 + 2 * half + 0) * DENC + n0 + mr];
        bb.y = Bb[(size_t)(k0 + 2 * half + 1) * DENC + n0 + mr];
        acc = wmma_f32(a, bb, acc);
    }
#pragma unroll
    for (int j = 0; j < 8; ++j) {
        int m = mTile * 16 + half * 8 + j;
        ctx[((size_t)b * S_ + m) * DENC + n0 + mr] = acc[j];
    }
}

// ---------------------------------------------------------------
// K4: hp[s,v] = hidden[s,:]·fcW[v,0:512] + fcb[v]   (32x32000)
// ---------------------------------------------------------------
__global__ __launch_bounds__(256) void fc_hidden_kernel(
    const float* __restrict__ hid, const float* __restrict__ W,
    const float* __restrict__ bvec, float* __restrict__ hp)
{
    const int wave  = blockIdx.x * 8 + (threadIdx.x >> 5);
    const int nTile = wave % (V_ / 16);   // 2000
    const int mTile = wave / (V_ / 16);   // 0..1
    const int t = threadIdx.x & 31;
    const int half = t >> 4, mr = t & 15;
    const int n0 = nTile * 16;

    const float* arow = hid + (size_t)(mTile * 16 + mr) * H_;
    const float* wrow = W + (size_t)(n0 + mr) * (H_ + DENC);   // K offset 0
    v8f acc = {};
#pragma unroll 4
    for (int k0 = 0; k0 < H_; k0 += 4) {
        v2f a = *(const v2f*)(arow + k0 + 2 * half);
        v2f b = *(const v2f*)(wrow + k0 + 2 * half);
        acc = wmma_f32(a, b, acc);
    }
    const float bn = bvec[n0 + mr];
#pragma unroll
    for (int j = 0; j < 8; ++j) {
        int m = mTile * 16 + half * 8 + j;
        hp[(size_t)m * V_ + n0 + mr] = acc[j] + bn;
    }
}

// ---------------------------------------------------------------
// K5: logits[m,v] = ctx[m,:]·fcW[v,512:1024] + hp[m%32, v]
//     LDS-tiled GEMM: block tile 128Mx128N, K chunked by 16,
//     double-buffered async global->LDS staging (ASYNCcnt).
//     8 waves/block, each wave computes 64x32 via 8 fp32 WMMAs/k-step.
// ---------------------------------------------------------------
constexpr int KC   = 16;   // k-chunk
constexpr int PAD  = 20;   // padded LDS row (floats): 16B-aligned, conflict-free
constexpr int MBLK = 128;
constexpr int NBLK = 128;

__global__ __launch_bounds__(256) void fc_main_kernel(
    const float* __restrict__ ctx, const float* __restrict__ W,
    const float* __restrict__ hp,  float* __restrict__ out)
{
    __shared__ float As[2][MBLK * PAD];   // 2 x 10 KB
    __shared__ float Bs[2][NBLK * PAD];   // 2 x 10 KB

    const int nb = blockIdx.x % (V_ / NBLK);   // 250
    const int mb = blockIdx.x / (V_ / NBLK);   // 8
    const int mBlk = mb * MBLK, nBlk = nb * NBLK;
    const int tid = threadIdx.x;
    const int w  = tid >> 5;
    const int wm = w >> 2;        // 0..1 -> 64 M rows
    const int wn = w & 3;         // 0..3 -> 32 N cols
    const int t = tid & 31, half = t >> 4, mr = t & 15;

    // staging assignment: 512 float4s per buffer, 2 per thread
    const int r0 = tid >> 2,          q0 = tid & 3;
    const int r1 = (tid + 256) >> 2,  q1 = tid & 3;
    const float* gA0 = ctx + (size_t)(mBlk + r0) * DENC + q0 * 4;
    const float* gA1 = ctx + (size_t)(mBlk + r1) * DENC + q1 * 4;
    const float* gB0 = W + (size_t)(nBlk + r0) * (H_ + DENC) + H_ + q0 * 4;
    const float* gB1 = W + (size_t)(nBlk + r1) * (H_ + DENC) + H_ + q1 * 4;
    const unsigned lA0 = (unsigned)(uintptr_t)&As[0][r0 * PAD + q0 * 4];
    const unsigned lA1 = (unsigned)(uintptr_t)&As[0][r1 * PAD + q1 * 4];
    const unsigned lB0 = (unsigned)(uintptr_t)&Bs[0][r0 * PAD + q0 * 4];
    const unsigned lB1 = (unsigned)(uintptr_t)&Bs[0][r1 * PAD + q1 * 4];
    const unsigned bufBytes = MBLK * PAD * 4;

    v8f acc[4][2] = {};

    // prologue: stage chunk 0 into buffer 0
    async_b128(lA0, gA0);
    async_b128(lA1, gA1);
    async_b128(lB0, gB0);
    async_b128(lB1, gB1);

    constexpr int NCHUNK = DENC / KC;   // 32
    for (int c = 0; c < NCHUNK; ++c) {
        wait_async0();
        __syncthreads();                 // chunk c visible to all waves

        if (c + 1 < NCHUNK) {            // stage chunk c+1 into other buffer
            const int k0 = (c + 1) * KC;
            const unsigned bo = ((c + 1) & 1) * bufBytes;
            async_b128(lA0 + bo, gA0 + k0);
            async_b128(lA1 + bo, gA1 + k0);
            async_b128(lB0 + bo, gB0 + k0);
            async_b128(lB1 + bo, gB1 + k0);
        }

        const float* As_b = &As[c & 1][0];
        const float* Bs_b = &Bs[c & 1][0];
#pragma unroll
        for (int kk = 0; kk < KC; kk += 4) {
            v2f af[4], bf[2];
#pragma unroll
            for (int i = 0; i < 4; ++i)
                af[i] = *(const v2f*)(As_b + (wm * 64 + i * 16 + mr) * PAD + kk + 2 * half);
#pragma unroll
            for (int j = 0; j < 2; ++j)
                bf[j] = *(const v2f*)(Bs_b + (wn * 32 + j * 16 + mr) * PAD + kk + 2 * half);
#pragma unroll
            for (int i = 0; i < 4; ++i)
#pragma unroll
                for (int j = 0; j < 2; ++j)
                    acc[i][j] = wmma_f32(af[i], bf[j], acc[i][j]);
        }
        __syncthreads();                 // all waves done with buffer (c&1)
    }

#pragma unroll
    for (int i = 0; i < 4; ++i) {
#pragma unroll
        for (int j = 0; j < 2; ++j) {
            const int n = nBlk + wn * 32 + j * 16 + mr;
#pragma unroll
            for (int jj = 0; jj < 8; ++jj) {
                const int m = mBlk + wm * 64 + i * 16 + half * 8 + jj;
                out[(size_t)m * V_ + n] = acc[i][j][jj] + hp[(size_t)(m & 31) * V_ + n];
            }
        }
    }
}

// ---------------------------------------------------------------
extern "C" void kernel_launch(void* const* d_in, const int* in_sizes, int n_in,
                              void* d_out, int out_size, void* d_ws, size_t ws_size,
                              hipStream_t stream) {
    const float* enc_out = (const float*)d_in[0];
    /* d_in[1] = target_seq (int64) — unused by the reference forward pass */
    const float* pos = (const float*)d_in[2];
    const float* dW  = (const float*)d_in[3];
    const float* db  = (const float*)d_in[4];
    const float* eW  = (const float*)d_in[5];
    const float* eb  = (const float*)d_in[6];
    const float* aW  = (const float*)d_in[7];
    const float* ab  = (const float*)d_in[8];
    const float* hW  = (const float*)d_in[9];
    const float* hb  = (const float*)d_in[10];
    const float* fW  = (const float*)d_in[11];
    const float* fb  = (const float*)d_in[12];
    float* out = (float*)d_out;

    // workspace layout (floats)
    float* ws   = (float*)d_ws;
    float* dq   = ws;                         // 32*256
    float* hid  = dq + S_ * A_;               // 32*512
    float* encp = hid + S_ * H_;              // 16384*256
    float* attn = encp + (size_t)M1 * A_;     // 32*32*512 (scores -> softmax in place)
    float* ctx  = attn + (size_t)M3 * L_;     // 32*32*512
    float* hp   = ctx + (size_t)M3 * DENC;    // 32*32000

    small_proj_kernel<<<96, 256, 0, stream>>>(pos, dW, db, hW, hb, dq, hid);
    enc_proj_kernel<<<(M1 / 16) * (A_ / 16) / 8, 256, 0, stream>>>(enc_out, eW, eb, encp);
    score_kernel<<<B_ * (L_ / 256), 256, 0, stream>>>(encp, dq, aW, ab, attn);
    softmax_kernel<<<B_ * S_, 256, 0, stream>>>(attn);
    context_kernel<<<B_ * 2 * (DENC / 16) / 8, 256, 0, stream>>>(attn, enc_out, ctx);
    fc_hidden_kernel<<<2 * (V_ / 16) / 8, 256, 0, stream>>>(hid, fW, fb, hp);
    fc_main_kernel<<<(M3 / MBLK) * (V_ / NBLK), 256, 0, stream>>>(ctx, fW, hp, out);
}